// GATLayer_88862873354914
// MI455X (gfx1250) — compile-verified
//
#include <hip/hip_runtime.h>

typedef __attribute__((ext_vector_type(2))) float v2f;
typedef __attribute__((ext_vector_type(8))) float v8f;

#define HEADS 8
#define IN_F  256
#define OUT_F 64
#define NN    4096

// ---------------------------------------------------------------------------
// Kernel 1: hp[h] = h (4096x256) @ W[h] (256x64), f32 WMMA 16x16x4.
// One wave32 computes a full 16x64 strip (4 accumulators): the A fragment is
// loaded once per K-step and reused by 4 consecutive WMMAs.
// A layout (16x4 f32): lanes 0-15 rows M=0..15 hold {K,K+1}; lanes 16-31 hold
// {K+2,K+3}.  B layout (4x16 f32): VGPR0 = rows K / K+2, VGPR1 = rows K+1/K+3.
// C/D (16x16 f32): VGPR r -> row m0+r (lanes 0-15) / m0+r+8 (lanes 16-31).
// ---------------------------------------------------------------------------
__global__ __launch_bounds__(256) void gat_hp_wmma(const float* __restrict__ h,
                                                   const float* __restrict__ W,
                                                   float* __restrict__ hp) {
    const int lane = threadIdx.x & 31;
    const int wave = threadIdx.x >> 5;
    const int tile = blockIdx.x * 8 + wave;     // 2048 strips: head*256 + mtile
    const int head = tile >> 8;
    const int m0   = (tile & 255) << 4;

    const int half = lane >> 4;                 // 0 -> {k,k+1}, 1 -> {k+2,k+3}
    const int l15  = lane & 15;

    const float* arow  = h + (m0 + l15) * IN_F + 2 * half;
    const float* bbase = W + head * (IN_F * OUT_F) + (2 * half) * OUT_F + l15;

    v8f c0 = {}, c1 = {}, c2 = {}, c3 = {};
#pragma unroll 4
    for (int k = 0; k < IN_F; k += 4) {
        v2f a = *reinterpret_cast<const v2f*>(arow + k);   // global_load_b64
        const float* b0 = bbase + k * OUT_F;
        v2f bA = { b0[0],  b0[OUT_F] };
        v2f bB = { b0[16], b0[16 + OUT_F] };
        v2f bC = { b0[32], b0[32 + OUT_F] };
        v2f bD = { b0[48], b0[48 + OUT_F] };
        c0 = __builtin_amdgcn_wmma_f32_16x16x4_f32(false, a, false, bA, (short)0, c0, false, false);
        c1 = __builtin_amdgcn_wmma_f32_16x16x4_f32(false, a, false, bB, (short)0, c1, false, false);
        c2 = __builtin_amdgcn_wmma_f32_16x16x4_f32(false, a, false, bC, (short)0, c2, false, false);
        c3 = __builtin_amdgcn_wmma_f32_16x16x4_f32(false, a, false, bD, (short)0, c3, false, false);
    }

    float* crow = hp + (size_t)(head * NN + m0 + 8 * half) * OUT_F + l15;
#pragma unroll
    for (int r = 0; r < 8; ++r) {
        crow[(size_t)r * OUT_F +  0] = c0[r];
        crow[(size_t)r * OUT_F + 16] = c1[r];
        crow[(size_t)r * OUT_F + 32] = c2[r];
        crow[(size_t)r * OUT_F + 48] = c3[r];
    }
}

// ---------------------------------------------------------------------------
// Kernel 2: e_dst[h,n] = hp[h,n,:] . a2[h,:]   (a1 / e_src cancel in softmax)
// float4 loads -> global_load_b128.
// ---------------------------------------------------------------------------
__global__ __launch_bounds__(256) void gat_edst(const float* __restrict__ hp,
                                                const float* __restrict__ a,
                                                float* __restrict__ e) {
    const int idx  = blockIdx.x * 256 + threadIdx.x;   // h*N + n
    const int head = idx >> 12;
    const float4* row4 = reinterpret_cast<const float4*>(hp + (size_t)idx * OUT_F);
    const float4* a24  = reinterpret_cast<const float4*>(a + head * (2 * OUT_F) + OUT_F);
    float s = 0.f;
#pragma unroll
    for (int o = 0; o < OUT_F / 4; ++o) {
        float4 r = row4[o], q = a24[o];
        s += r.x * q.x + r.y * q.y + r.z * q.z + r.w * q.w;
    }
    e[idx] = s;
}

// ---------------------------------------------------------------------------
// Kernel 3: per-head softmax over N (in place on e).
// ---------------------------------------------------------------------------
__global__ __launch_bounds__(256) void gat_softmax(float* __restrict__ e) {
    __shared__ float red[256];
    float* eh   = e + blockIdx.x * NN;
    const int t = threadIdx.x;

    float m = -3.402823466e+38f;
    for (int j = t; j < NN; j += 256) m = fmaxf(m, eh[j]);
    red[t] = m; __syncthreads();
    for (int s = 128; s > 0; s >>= 1) {
        if (t < s) red[t] = fmaxf(red[t], red[t + s]);
        __syncthreads();
    }
    m = red[0]; __syncthreads();

    float sum = 0.f;
    for (int j = t; j < NN; j += 256) sum += __expf(eh[j] - m);
    red[t] = sum; __syncthreads();
    for (int s = 128; s > 0; s >>= 1) {
        if (t < s) red[t] += red[t + s];
        __syncthreads();
    }
    const float inv = 1.0f / red[0];

    for (int j = t; j < NN; j += 256) eh[j] = __expf(eh[j] - m) * inv;
}

// ---------------------------------------------------------------------------
// Kernel 4: out_row[h,o] = sum_j p[h,j] * hp[h,j,o]
// ---------------------------------------------------------------------------
__global__ __launch_bounds__(256) void gat_reduce(const float* __restrict__ hp,
                                                  const float* __restrict__ p,
                                                  float* __restrict__ out_row) {
    __shared__ float red[256];
    const int head = blockIdx.x;
    const int t  = threadIdx.x;
    const int o  = t & 63;
    const int js = t >> 6;
    const float* ph  = p + head * NN;
    const float* hph = hp + (size_t)head * NN * OUT_F + o;
    float acc = 0.f;
    for (int j = js; j < NN; j += 4)
        acc += ph[j] * hph[(size_t)j * OUT_F];
    red[t] = acc; __syncthreads();
    if (t < 64)
        out_row[head * OUT_F + t] = red[t] + red[t + 64] + red[t + 128] + red[t + 192];
}

// ---------------------------------------------------------------------------
// Kernel 5: broadcast out_row (512 floats) to every output row; b128 stores.
// ---------------------------------------------------------------------------
__global__ __launch_bounds__(256) void gat_bcast(const float* __restrict__ out_row,
                                                 float* __restrict__ out) {
    __shared__ float s[512];
    if (threadIdx.x < 256) {
        s[threadIdx.x]       = out_row[threadIdx.x];
        s[threadIdx.x + 256] = out_row[threadIdx.x + 256];
    }
    __syncthreads();
    float4* out4 = reinterpret_cast<float4*>(out);
    const float4* s4 = reinterpret_cast<const float4*>(s);
    const size_t base = (size_t)blockIdx.x * 1024;   // 4096 floats per block
    for (int i = threadIdx.x; i < 1024; i += 256)
        out4[base + i] = s4[i & 127];
}

extern "C" void kernel_launch(void* const* d_in, const int* in_sizes, int n_in,
                              void* d_out, int out_size, void* d_ws, size_t ws_size,
                              hipStream_t stream) {
    const float* h   = (const float*)d_in[0];
    const float* adj = (const float*)d_in[1];  (void)adj;  // unused by reference
    const float* W   = (const float*)d_in[2];
    const float* a   = (const float*)d_in[3];
    float* out = (float*)d_out;

    float* hp      = (float*)d_ws;                 // 8*4096*64 = 2,097,152 f32
    float* e       = hp + (size_t)HEADS * NN * OUT_F;
    float* out_row = e + (size_t)HEADS * NN;       // 512 f32

    gat_hp_wmma<<<256, 256, 0, stream>>>(h, W, hp);
    gat_edst  <<<(HEADS * NN) / 256, 256, 0, stream>>>(hp, a, e);
    gat_softmax<<<HEADS, 256, 0, stream>>>(e);
    gat_reduce<<<HEADS, 256, 0, stream>>>(hp, e, out_row);
    gat_bcast <<<512, 256, 0, stream>>>(out_row, out);
}